// GraphSAGE_79834852098094
// MI455X (gfx1250) — compile-verified
//
#include <hip/hip_runtime.h>

// GraphSAGE (2x SAGEConv + global mean pool) for MI455X / gfx1250.
//
// Roofline reasoning:
//  - Edge aggregation dominates: 2 layers x (3.2M edges x 128ch) gathers and
//    f32 atomic adds. Node features = 100000*128*4B = 51 MB -> fully resident
//    in the 192 MB L2, so gather+scatter run at L2 speed, not HBM.
//    Scatter uses one wave per edge with float4 lanes: 32x global_load_b128
//    + 128x non-returning global_atomic_add_f32 per edge.
//  - GEMMs are ~10 GFLOP total -> tiny. Use the full-precision f32 WMMA
//    (V_WMMA_F32_16X16X4_F32) so matrix math matches the fp32 reference;
//    f16/bf16 WMMA rate is not needed since we're memory bound.

#define N_NODES   100000
#define N_EDGES   3200000
#define IN_C      128
#define HID_C     128
#define OUT_C     64
#define NUM_GRAPHS 1024

typedef float v2f __attribute__((ext_vector_type(2)));
typedef float v8f __attribute__((ext_vector_type(8)));

// ---- degree of each destination node (shared by both layers) ----
__global__ void deg_kernel(const int* __restrict__ dst, float* __restrict__ degf) {
    int e = blockIdx.x * blockDim.x + threadIdx.x;
    if (e < N_EDGES) atomicAdd(&degf[dst[e]], 1.0f);
}

// ---- nodes per graph (for global_mean_pool) ----
__global__ void gcnt_kernel(const int* __restrict__ batch, float* __restrict__ gcnt) {
    int i = blockIdx.x * blockDim.x + threadIdx.x;
    if (i < N_NODES) atomicAdd(&gcnt[batch[i]], 1.0f);
}

// ---- scatter-add of 128-wide features along edges: acc[dst] += feat[src] ----
// One wave (32 lanes) per edge, float4 per lane: the 512B source row comes in
// as coalesced global_load_b128; 4 non-returning global_atomic_add_f32 per
// lane (STOREcnt -> no wave stall). e is wave-uniform so the index loads
// scalarize. 8 edges per 256-thread block.
__global__ __launch_bounds__(256)
void scatter128_kernel(const int* __restrict__ src, const int* __restrict__ dst,
                       const float* __restrict__ feat, float* __restrict__ acc) {
    int e    = blockIdx.x * 8 + (threadIdx.x >> 5);
    int lane = threadIdx.x & 31;
    if (e < N_EDGES) {
        int s = src[e];
        int d = dst[e];
        const float4* fp = (const float4*)(feat + (size_t)s * 128);
        float4 v = fp[lane];
        float* ap = acc + (size_t)d * 128 + lane * 4;
        atomicAdd(ap + 0, v.x);
        atomicAdd(ap + 1, v.y);
        atomicAdd(ap + 2, v.z);
        atomicAdd(ap + 3, v.w);
    }
}

// ---- fused SAGE layer GEMM:
//   out[i] = act( (nbsum[i]/max(deg[i],1)) @ Wl  +  bias  +  selfx[i] @ Wr )
// K is always 128 (IN_C == HID_C). NC = output cols (128 or 64).
// One block per 16-row node tile; one wave per 16-col output tile.
// f32 WMMA 16x16x4: A/B are 2 VGPRs/lane, C/D 8 VGPRs/lane.
template<int NC, bool RELU>
__global__ __launch_bounds__(NC * 2)
void sage_gemm_kernel(const float* __restrict__ nbsum, const float* __restrict__ degf,
                      const float* __restrict__ selfx, const float* __restrict__ Wl,
                      const float* __restrict__ bias,  const float* __restrict__ Wr,
                      float* __restrict__ out) {
    // stride 132 (132 % 64 == 4) -> the column-broadcast A reads
    // (lanes 0..15 hit rows 0..15 at the same k) land in distinct banks.
    __shared__ float ldsMean[16][132];
    __shared__ float ldsX[16][132];
    __shared__ float sscale[16];

    const int t = threadIdx.x;
    const int rowbase = blockIdx.x * 16;          // N_NODES == 6250*16 exactly

    if (t < 16) sscale[t] = 1.0f / fmaxf(degf[rowbase + t], 1.0f);
    __syncthreads();

    const int nthreads = NC * 2;
    for (int idx = t; idx < 16 * 128; idx += nthreads) {
        int r  = idx >> 7;
        int cc = idx & 127;
        long o = (long)(rowbase + r) * 128 + cc;
        ldsMean[r][cc] = nbsum[o] * sscale[r];
        ldsX[r][cc]    = selfx[o];
    }
    __syncthreads();

    const int wave    = t >> 5;
    const int lane    = t & 31;
    const int colbase = wave * 16;
    const int arow    = lane & 15;                // A: M index
    const int kb      = (lane >> 4) * 2;          // A/B: K sub-offset (0 or 2)
    const int bcol    = lane & 15;                // B: N index

    v8f c = {};
    #pragma unroll 4
    for (int k0 = 0; k0 < 128; k0 += 4) {
        v2f a, b;
        // mean-neighbor term: A = ldsMean tile, B = Wl[k, col]
        a[0] = ldsMean[arow][k0 + kb];
        a[1] = ldsMean[arow][k0 + kb + 1];
        b[0] = Wl[(k0 + kb)     * NC + colbase + bcol];
        b[1] = Wl[(k0 + kb + 1) * NC + colbase + bcol];
        c = __builtin_amdgcn_wmma_f32_16x16x4_f32(false, a, false, b,
                                                  (short)0, c, false, false);
        // self term: A = ldsX tile, B = Wr[k, col]
        a[0] = ldsX[arow][k0 + kb];
        a[1] = ldsX[arow][k0 + kb + 1];
        b[0] = Wr[(k0 + kb)     * NC + colbase + bcol];
        b[1] = Wr[(k0 + kb + 1) * NC + colbase + bcol];
        c = __builtin_amdgcn_wmma_f32_16x16x4_f32(false, a, false, b,
                                                  (short)0, c, false, false);
    }

    const float bv = bias[colbase + bcol];
    const int rowoff = (lane < 16) ? 0 : 8;       // D: vgpr r -> row r or r+8
    #pragma unroll
    for (int r = 0; r < 8; ++r) {
        float v = c[r] + bv;
        if (RELU) v = fmaxf(v, 0.0f);
        out[(long)(rowbase + r + rowoff) * NC + colbase + bcol] = v;
    }
}

// ---- global mean pool: accumulate then divide ----
__global__ void pool_kernel(const int* __restrict__ batch, const float* __restrict__ h2,
                            float* __restrict__ outp) {
    int idx = blockIdx.x * blockDim.x + threadIdx.x;
    if (idx < N_NODES * OUT_C) {
        int i = idx >> 6;
        int c = idx & 63;
        atomicAdd(&outp[batch[i] * OUT_C + c], h2[idx]);
    }
}

__global__ void div_kernel(const float* __restrict__ gcnt, float* __restrict__ outp) {
    int idx = blockIdx.x * blockDim.x + threadIdx.x;
    if (idx < NUM_GRAPHS * OUT_C) {
        outp[idx] = outp[idx] / fmaxf(gcnt[idx >> 6], 1.0f);
    }
}

extern "C" void kernel_launch(void* const* d_in, const int* in_sizes, int n_in,
                              void* d_out, int out_size, void* d_ws, size_t ws_size,
                              hipStream_t stream) {
    // setup_inputs() dict order:
    const float* x    = (const float*)d_in[0];   // [N, 128]
    const float* W1l  = (const float*)d_in[1];   // [128, 128]
    const float* b1   = (const float*)d_in[2];   // [128]
    const float* W1r  = (const float*)d_in[3];   // [128, 128]
    const float* W2l  = (const float*)d_in[4];   // [128, 64]
    const float* b2   = (const float*)d_in[5];   // [64]
    const float* W2r  = (const float*)d_in[6];   // [128, 64]
    const int*   ei   = (const int*)d_in[7];     // [2, E]
    const int*   batch= (const int*)d_in[8];     // [N]
    const int* src = ei;
    const int* dst = ei + N_EDGES;
    float* outp = (float*)d_out;                 // [1024, 64]

    // workspace carving (256B aligned)
    char* ws = (char*)d_ws;
    size_t off = 0;
    auto carve = [&](size_t bytes) -> void* {
        void* p = ws + off;
        off = (off + bytes + 255) & ~(size_t)255;
        return p;
    };
    float* degf = (float*)carve((size_t)N_NODES * 4);
    float* gcnt = (float*)carve((size_t)NUM_GRAPHS * 4);
    float* sum  = (float*)carve((size_t)N_NODES * 128 * 4);  // reused by both layers
    float* h    = (float*)carve((size_t)N_NODES * 128 * 4);
    float* h2   = (float*)carve((size_t)N_NODES * 64 * 4);

    // zero accumulators (graph-capturable memset nodes)
    hipMemsetAsync(degf, 0, (size_t)N_NODES * 4, stream);
    hipMemsetAsync(gcnt, 0, (size_t)NUM_GRAPHS * 4, stream);
    hipMemsetAsync(sum,  0, (size_t)N_NODES * 128 * 4, stream);
    hipMemsetAsync(d_out, 0, (size_t)NUM_GRAPHS * OUT_C * 4, stream);

    deg_kernel <<<(N_EDGES + 255) / 256, 256, 0, stream>>>(dst, degf);
    gcnt_kernel<<<(N_NODES + 255) / 256, 256, 0, stream>>>(batch, gcnt);

    // layer 1
    scatter128_kernel<<<(N_EDGES + 7) / 8, 256, 0, stream>>>(src, dst, x, sum);
    sage_gemm_kernel<128, true><<<N_NODES / 16, 256, 0, stream>>>(
        sum, degf, x, W1l, b1, W1r, h);

    // layer 2 (reuse sum buffer)
    hipMemsetAsync(sum, 0, (size_t)N_NODES * 128 * 4, stream);
    scatter128_kernel<<<(N_EDGES + 7) / 8, 256, 0, stream>>>(src, dst, h, sum);
    sage_gemm_kernel<64, false><<<N_NODES / 16, 128, 0, stream>>>(
        sum, degf, h, W2l, b2, W2r, h2);

    // global mean pool
    pool_kernel<<<(N_NODES * OUT_C + 255) / 256, 256, 0, stream>>>(batch, h2, outp);
    div_kernel <<<(NUM_GRAPHS * OUT_C + 255) / 256, 256, 0, stream>>>(gcnt, outp);
}